// TranVectorQuantizer_65292092834255
// MI455X (gfx1250) — compile-verified
//
#include <hip/hip_runtime.h>

typedef __attribute__((ext_vector_type(2))) float v2f;
typedef __attribute__((ext_vector_type(4))) float v4f;
typedef __attribute__((ext_vector_type(8))) float v8f;

#define KCODES 128
#define DIM    32
#define NROWS  (32768 * 8)          // 262144 flat vectors
#define OUT1   (NROWS * DIM)        // 8,388,608 floats per latent-shaped output

// ---------------------------------------------------------------------------
// Kernel 1: argmin over codebook via f32 WMMA + gather of quantized vectors.
// One wave handles a 16-row tile. score(k) = ||c_k||^2 - 2 x.c_k  (x-norm is
// constant per row -> irrelevant for argmin). Codebook pre-scaled by -2 and
// norms precomputed into LDS; WMMA chain computes the full score directly.
// ---------------------------------------------------------------------------
__global__ __launch_bounds__(256) void vq_wmma_kernel(
    const float* __restrict__ latent,
    const float* __restrict__ codebook,
    float* __restrict__ out_policy,
    float* __restrict__ out_quant)
{
    __shared__ float cb_s[KCODES * DIM];   // -2 * codebook, row-major [k][d]
    __shared__ float cnorm[KCODES];        // ||c_k||^2

    const int tid = threadIdx.x;

    // Stage codebook (scaled by -2) into LDS.
    for (int i = tid; i < KCODES * DIM; i += blockDim.x)
        cb_s[i] = -2.0f * codebook[i];
    // Per-code squared norms.
    if (tid < KCODES) {
        float s = 0.0f;
        #pragma unroll
        for (int d = 0; d < DIM; ++d) {
            float v = codebook[tid * DIM + d];
            s += v * v;
        }
        cnorm[tid] = s;
    }
    __syncthreads();

    const int wave    = tid >> 5;                 // 0..7
    const int lane    = tid & 31;
    const int half    = lane >> 4;                // 0: lanes 0-15, 1: lanes 16-31
    const int lm      = lane & 15;                // column / row-within-half id
    const int rowBase = (blockIdx.x * 8 + wave) * 16;
    if (rowBase >= NROWS) return;                 // wave-uniform (EXEC stays all-1)

    // --- Load A fragments: 16x32 rows as eight 16x4 f32 WMMA A-tiles.
    // ISA layout (32-bit A 16x4): lanes 0-15 -> K={0,1}, lanes 16-31 -> K={2,3}.
    const int arow = rowBase + lm;
    v2f a[8];
    #pragma unroll
    for (int c = 0; c < 8; ++c) {
        const int k0 = 4 * c + 2 * half;
        a[c] = *(const v2f*)(latent + (size_t)arow * DIM + k0);
    }

    // Running per-lane argmin: VGPR r of the accumulator holds row
    // M = r (half 0) or M = 8 + r (half 1); column N = n0 + lm.
    float bestV[8];
    int   bestI[8];
    #pragma unroll
    for (int r = 0; r < 8; ++r) { bestV[r] = 3.0e38f; bestI[r] = 0; }

    #pragma unroll
    for (int t = 0; t < 8; ++t) {                 // 8 code tiles of 16 codes
        const int n0  = t * 16;
        const int col = n0 + lm;

        const float cn = cnorm[col];
        v8f acc;
        #pragma unroll
        for (int r = 0; r < 8; ++r) acc[r] = cn;  // seed C with ||c||^2

        #pragma unroll
        for (int c = 0; c < 8; ++c) {             // chain K=32 as 8x (16x16x4)
            const int k0 = 4 * c + 2 * half;
            v2f b = *(const v2f*)(cb_s + col * DIM + k0);
            acc = __builtin_amdgcn_wmma_f32_16x16x4_f32(
                false, a[c], false, b, (short)0, acc, false, false);
        }

        #pragma unroll
        for (int r = 0; r < 8; ++r) {
            float v = acc[r];
            if (v < bestV[r] || (v == bestV[r] && col < bestI[r])) {
                bestV[r] = v;
                bestI[r] = col;
            }
        }
    }

    // Butterfly min-reduce across the 16 columns held in each 16-lane half.
    #pragma unroll
    for (int m = 1; m <= 8; m <<= 1) {
        #pragma unroll
        for (int r = 0; r < 8; ++r) {
            float ov = __shfl_xor(bestV[r], m, 32);
            int   oi = __shfl_xor(bestI[r], m, 32);
            if (ov < bestV[r] || (ov == bestV[r] && oi < bestI[r])) {
                bestV[r] = ov;
                bestI[r] = oi;
            }
        }
    }

    // Gather + store: row m (0..7 from half 0, 8..15 from half 1). Each lane
    // writes one of the 32 dims; LDS holds -2*c so multiply by -0.5 (exact).
    #pragma unroll
    for (int m = 0; m < 16; ++m) {
        const int r   = m & 7;
        const int src = (m < 8) ? 0 : 16;
        const int idx = __shfl(bestI[r], src, 32);
        const float val = -0.5f * cb_s[idx * DIM + lane];
        const size_t o = (size_t)(rowBase + m) * DIM + lane;
        __builtin_nontemporal_store(val, out_policy + o);
        __builtin_nontemporal_store(val, out_quant  + o);
    }
}

// ---------------------------------------------------------------------------
// Kernel 2: codebook_set broadcast — 512 MB of writes, pure bandwidth.
// Stage the 16 KB codebook in LDS, then grid-stride float4 NT stores.
// ---------------------------------------------------------------------------
__global__ __launch_bounds__(256) void cbset_bcast_kernel(
    const float* __restrict__ codebook,
    float* __restrict__ out_cbset)
{
    __shared__ v4f cb4[KCODES * DIM / 4];         // 1024 float4

    for (int i = threadIdx.x; i < KCODES * DIM / 4; i += blockDim.x)
        cb4[i] = *(const v4f*)(codebook + 4 * i);
    __syncthreads();

    v4f* __restrict__ out4 = (v4f*)out_cbset;
    const size_t total  = (size_t)32768 * (KCODES * DIM / 4);  // 33,554,432
    const size_t stride = (size_t)gridDim.x * blockDim.x;
    for (size_t i = (size_t)blockIdx.x * blockDim.x + threadIdx.x;
         i < total; i += stride) {
        v4f v = cb4[i & 1023];                    // K*D/4 = 1024 (pow2)
        __builtin_nontemporal_store(v, out4 + i);
    }
}

extern "C" void kernel_launch(void* const* d_in, const int* in_sizes, int n_in,
                              void* d_out, int out_size, void* d_ws, size_t ws_size,
                              hipStream_t stream)
{
    const float* latent   = (const float*)d_in[0];   // [32768, 8, 32] f32
    const float* codebook = (const float*)d_in[1];   // [128, 32] f32

    float* out        = (float*)d_out;
    float* out_policy = out;                 // [32768, 8, 32]
    float* out_quant  = out + OUT1;          // [32768, 8, 32]
    float* out_cbset  = out + 2 * OUT1;      // [32768, 128, 32]

    // 262144 rows / 16 rows-per-wave / 8 waves-per-block = 2048 blocks.
    vq_wmma_kernel<<<2048, 256, 0, stream>>>(latent, codebook, out_policy, out_quant);

    // 33.5M float4 stores; 8192 blocks x 256 threads -> 16 float4 per thread.
    cbset_bcast_kernel<<<8192, 256, 0, stream>>>(codebook, out_cbset);
}